// Eq2Net_7859790151694
// MI455X (gfx1250) — compile-verified
//
#include <hip/hip_runtime.h>
#include <hip/hip_bf16.h>
#include <math.h>

#define NBLK   32      // B_ABS
#define NACT   128     // N_ACT
#define SDIM   512
#define HIDD   64
#define TLEN   1024
#define TP1    1025
#define PENV   0.5f
#define NEGV   -1e30f
#define KC     64      // GEMM K-chunk staged per TDM descriptor pair

typedef __attribute__((ext_vector_type(2))) float v2f;
typedef __attribute__((ext_vector_type(8))) float v8f;
typedef __attribute__((ext_vector_type(4))) unsigned int v4u;
typedef __attribute__((ext_vector_type(4))) int v4i;
typedef __attribute__((ext_vector_type(8))) int v8i;

#if __has_builtin(__builtin_amdgcn_tensor_load_to_lds) && \
    __has_builtin(__builtin_amdgcn_s_wait_tensorcnt)
#define USE_TDM 1
#else
#define USE_TDM 0
#endif

__device__ __forceinline__ float wave_max(float v) {
#pragma unroll
  for (int o = 16; o > 0; o >>= 1) v = fmaxf(v, __shfl_xor(v, o, 32));
  return v;
}
__device__ __forceinline__ float wave_sum(float v) {
#pragma unroll
  for (int o = 16; o > 0; o >>= 1) v += __shfl_xor(v, o, 32);
  return v;
}

#if USE_TDM
typedef __attribute__((address_space(3))) void lds_void;

__device__ __forceinline__ unsigned lds_addr_of(void* p) {
  return (unsigned)(unsigned long long)(lds_void*)p;  // AS3 ptrtoint = LDS byte offset
}

// Issue one 2D-tile TDM load: tile_d1 rows of tile_d0 f32 elements, row stride
// stride0 (elements).  tensor_d0/d1 bound the valid region (OOB reads zero).
// pad!=0 inserts 1 dword of LDS padding per 64 dwords (row stride 65 in LDS).
__device__ __forceinline__ void tdm_load_2d(const float* gp, unsigned lds_byte,
                                            unsigned tensor_d0, unsigned tensor_d1,
                                            unsigned tile_d0, unsigned tile_d1,
                                            unsigned long long stride0, int pad) {
  const unsigned long long ga = (unsigned long long)gp;
  v4u g0;
  g0[0] = 1u;                                            // count=1, user mode
  g0[1] = lds_byte;                                      // lds_addr
  g0[2] = (unsigned)(ga & 0xFFFFFFFFu);                  // global_addr[31:0]
  g0[3] = (unsigned)((ga >> 32) & 0x1FFFFFFu) | (2u << 30);  // addr[56:32] | type=2
  unsigned w0 = (2u << 16);                              // data_size = 4 bytes
  if (pad) w0 |= (1u << 20) | (5u << 22);                // pad_enable, 64dw interval, 1dw pad
  v8i g1;
  g1[0] = (int)w0;
  g1[1] = (int)((tensor_d0 & 0xFFFFu) << 16);            // tensor_dim0[15:0] @ bit48
  g1[2] = (int)((tensor_d0 >> 16) | ((tensor_d1 & 0xFFFFu) << 16));
  g1[3] = (int)((tensor_d1 >> 16) | (tile_d0 << 16));    // tile_dim0 @ bit112
  g1[4] = (int)(tile_d1 & 0xFFFFu);                      // tile_dim1, tile_dim2=0
  g1[5] = (int)(stride0 & 0xFFFFFFFFull);                // tensor_dim0_stride[31:0]
  g1[6] = (int)((stride0 >> 32) & 0xFFFFull);
  g1[7] = 0;
  const v4i z = {0, 0, 0, 0};
#if __clang_major__ >= 23
  const v8i z8 = {0, 0, 0, 0, 0, 0, 0, 0};
  __builtin_amdgcn_tensor_load_to_lds(g0, g1, z, z, z8, 0);
#else
  __builtin_amdgcn_tensor_load_to_lds(g0, g1, z, z, 0);
#endif
}
#endif  // USE_TDM

// ---------------------------------------------------------------------------
// LDS-tiled f32 WMMA GEMM: C[M,N] = opt_relu(A[M,K] @ B[K,N] + bias[N]).
// 8 waves (2x4) per block compute a 32x64 C tile; K in chunks of KC=64 staged
// into LDS by the Tensor Data Mover (double-buffered, TENSORcnt-tracked).
// A LDS tile uses TDM padding for a 65-float row stride (conflict-free reads).
// ---------------------------------------------------------------------------
__global__ __launch_bounds__(256) void gemm_wmma_f32(
    const float* __restrict__ A, const float* __restrict__ B,
    const float* __restrict__ bias, float* __restrict__ C,
    int M, int N, int K, int relu) {
  __shared__ float sA[2][32 * 65];
  __shared__ float sB[2][KC * 64];
  const int tid  = threadIdx.x;
  const int lane = tid & 31;
  const int wave = tid >> 5;
  const int mw = wave >> 2, nw = wave & 3;
  const int hi = lane >> 4, lo = lane & 15;
  const int m0 = blockIdx.y * 32;
  const int n0 = blockIdx.x * 64;
  const int nc = K / KC;

  const int col = n0 + 16 * nw + lo;
  const int colc = (col < N) ? col : (N - 1);
  const float bv = bias[colc];
  v8f acc;
#pragma unroll
  for (int r = 0; r < 8; ++r) acc[r] = bv;

#if USE_TDM
  const unsigned ldsA0 = lds_addr_of(&sA[0][0]);
  const unsigned ldsA1 = lds_addr_of(&sA[1][0]);
  const unsigned ldsB0 = lds_addr_of(&sB[0][0]);
  const unsigned ldsB1 = lds_addr_of(&sB[1][0]);
  if (tid < 32) {  // wave 0 issues chunk 0 (EXEC-independent scalar DMA)
    tdm_load_2d(A + m0 * K, ldsA0, (unsigned)K, (unsigned)(M - m0),
                KC, 32, (unsigned long long)K, 1);
    tdm_load_2d(B + n0, ldsB0, (unsigned)(N - n0), (unsigned)K,
                KC, KC, (unsigned long long)N, 0);
  }
#endif

  for (int c = 0; c < nc; ++c) {
    const int buf = c & 1;
#if USE_TDM
    if (c + 1 < nc) {
      const int k1 = (c + 1) * KC;
      if (tid < 32) {
        tdm_load_2d(A + m0 * K + k1, buf ? ldsA0 : ldsA1,
                    (unsigned)(K - k1), (unsigned)(M - m0),
                    KC, 32, (unsigned long long)K, 1);
        tdm_load_2d(B + k1 * N + n0, buf ? ldsB0 : ldsB1,
                    (unsigned)(N - n0), (unsigned)(K - k1),
                    KC, KC, (unsigned long long)N, 0);
        __builtin_amdgcn_s_wait_tensorcnt(2);  // oldest pair (chunk c) done
      }
    } else if (tid < 32) {
      __builtin_amdgcn_s_wait_tensorcnt(0);
    }
    __syncthreads();
    const float* a_base = &sA[buf][(16 * mw + lo) * 65];
    const float* b_base = &sB[buf][16 * nw + lo];
#else
    {  // cooperative staging fallback (single buffer)
      const int k0 = c * KC;
      for (int idx = tid; idx < 32 * KC; idx += 256) {
        const int r = idx / KC, cc = idx % KC;
        const int gr = m0 + r;
        sA[0][r * 65 + cc] = (gr < M) ? A[gr * K + k0 + cc] : 0.0f;
      }
      for (int idx = tid; idx < KC * 64; idx += 256) {
        const int r = idx / 64, cc = idx % 64;
        const int gc = n0 + cc;
        sB[0][r * 64 + cc] = (gc < N) ? B[(k0 + r) * N + gc] : 0.0f;
      }
    }
    __syncthreads();
    const float* a_base = &sA[0][(16 * mw + lo) * 65];
    const float* b_base = &sB[0][16 * nw + lo];
#endif

#pragma unroll
    for (int kk = 0; kk < KC; kk += 4) {
      const int ka = kk + 2 * hi;
      v2f a; a.x = a_base[ka];       a.y = a_base[ka + 1];
      v2f b; b.x = b_base[ka * 64];  b.y = b_base[(ka + 1) * 64];
      acc = __builtin_amdgcn_wmma_f32_16x16x4_f32(
          false, a, false, b, (short)0, acc, false, false);
    }
    __syncthreads();
  }

#pragma unroll
  for (int r = 0; r < 8; ++r) {
    const int orow = m0 + 16 * mw + r + 8 * hi;
    if (orow < M && col < N) {
      float v = acc[r];
      if (relu) v = fmaxf(v, 0.0f);
      C[orow * N + col] = v;
    }
  }
}

// step_logps[t,b] = logits[t,b,act[t]] - lse_a(logits[t,b,:]).  One wave per (t,b).
__global__ __launch_bounds__(128) void step_logps_kernel(
    const float* __restrict__ logits, const int* __restrict__ actions,
    float* __restrict__ step) {
  const int lane = threadIdx.x & 31;
  const int p = blockIdx.x * 4 + (threadIdx.x >> 5);  // < TLEN*NBLK
  const int t = p >> 5, b = p & 31;
  const float* rowp = logits + t * (NBLK * NACT) + b * NACT;
  const float v0 = rowp[lane], v1 = rowp[lane + 32];
  const float v2 = rowp[lane + 64], v3 = rowp[lane + 96];
  float m = fmaxf(fmaxf(v0, v1), fmaxf(v2, v3));
  m = wave_max(m);
  float s = __expf(v0 - m) + __expf(v1 - m) + __expf(v2 - m) + __expf(v3 - m);
  s = wave_sum(s);
  if (lane == 0) {
    const int a = actions[t];
    step[t * NBLK + b] = rowp[a] - (m + __logf(s));
  }
}

// stop log-softmax over pairs + start log-softmax over 32.  One wave per row t.
__global__ __launch_bounds__(128) void heads_kernel(
    const float* __restrict__ Zs, const float* __restrict__ Zg,
    float* __restrict__ stop0, float* __restrict__ stop1,
    float* __restrict__ startlp) {
  const int lane = threadIdx.x & 31;
  const int t = blockIdx.x * 4 + (threadIdx.x >> 5);
  if (t >= TP1) return;
  const float z0 = Zs[t * 64 + 2 * lane];
  const float z1 = Zs[t * 64 + 2 * lane + 1];
  const float m = fmaxf(z0, z1);
  const float l = m + __logf(__expf(z0 - m) + __expf(z1 - m));
  stop0[t * 32 + lane] = z0 - l;
  stop1[t * 32 + lane] = z1 - l;
  const float g = Zg[t * 32 + lane];
  const float gm = wave_max(g);
  const float gs = wave_sum(__expf(g - gm));
  startlp[t * 32 + lane] = g - (gm + __logf(gs));
}

// Serial prefix sums over t (one lane per b), fused with L/R assembly.
__global__ void scan_kernel(const float* __restrict__ step,
                            const float* __restrict__ stop0,
                            const float* __restrict__ stop1,
                            const float* __restrict__ startlp,
                            float* __restrict__ S, float* __restrict__ C0,
                            float* __restrict__ L, float* __restrict__ R) {
  const int b = threadIdx.x;  // 0..31
  float s = 0.f, c = 0.f;
  S[b] = 0.f;
  C0[b] = 0.f;
  for (int t = 0; t < TLEN; ++t) {
    R[t * 32 + b] = s + c + stop1[t * 32 + b];       // S[t] + C0[t] + stop1[t]
    float Lv = startlp[t * 32 + b] - s;              // - S[t]
    s += step[t * 32 + b];
    c += stop0[t * 32 + b];
    S[(t + 1) * 32 + b] = s;
    C0[(t + 1) * 32 + b] = c;
    L[t * 32 + b] = Lv - c;                          // - C0[t+1]
  }
  R[TLEN * 32 + b] = s + c + stop1[TLEN * 32 + b];
}

// A[i,j] = lse_b(L[i,b] + R[j,b]).  16x16 output tile per block, operands in LDS.
__global__ __launch_bounds__(256) void pairA_kernel(
    const float* __restrict__ L, const float* __restrict__ R,
    float* __restrict__ A) {
  __shared__ float Ls[16 * 32];
  __shared__ float Rs[16 * 32];
  const int tid = threadIdx.x;
  const int i0 = blockIdx.y * 16;
  const int j0 = blockIdx.x * 16;
  {
    const int r = tid >> 5, b = tid & 31;
    Ls[tid]       = L[(i0 + r) * 32 + b];
    Ls[tid + 256] = L[(i0 + r + 8) * 32 + b];
    Rs[tid]       = R[min(j0 + r, TLEN) * 32 + b];
    Rs[tid + 256] = R[min(j0 + r + 8, TLEN) * 32 + b];
  }
  __syncthreads();
  const int ti = tid >> 4, tj = tid & 15;
  const float* lp = Ls + ti * 32;
  const float* rp = Rs + tj * 32;
  float m = NEGV;
#pragma unroll
  for (int b = 0; b < 32; ++b) m = fmaxf(m, lp[b] + rp[b]);
  float sum = 0.f;
#pragma unroll
  for (int b = 0; b < 32; ++b) sum += __expf(lp[b] + rp[b] - m);
  const int j = j0 + tj;
  if (j < TP1) A[(i0 + ti) * TP1 + j] = m + __logf(sum);
}

// Sequential semiring triangular solve: table[i] = lse_{j>i}(A[i,j]+table[j]) - PEN.
// Single workgroup, table in LDS, two-phase (max,sum) block reductions.
__global__ __launch_bounds__(1024) void dp_kernel(const float* __restrict__ A,
                                                  float* __restrict__ out) {
  __shared__ float table[TP1];
  __shared__ float red[32];
  __shared__ float bcast[1];
  const int tid = threadIdx.x;
  const int lane = tid & 31, wid = tid >> 5;
  if (tid < TLEN) table[tid] = NEGV;
  if (tid == 0) table[TLEN] = 0.f;
  __syncthreads();
  for (int i = TLEN - 1; i >= 0; --i) {
    const int j = i + 1 + tid;
    const bool act = (j <= TLEN);
    const float v = act ? (A[i * TP1 + j] + table[j]) : NEGV;
    float m = wave_max(v);
    if (lane == 0) red[wid] = m;
    __syncthreads();
    if (wid == 0) {
      float mm = wave_max(red[lane]);
      if (lane == 0) bcast[0] = mm;
    }
    __syncthreads();
    const float M = bcast[0];
    float s = wave_sum(act ? __expf(v - M) : 0.f);
    if (lane == 0) red[wid] = s;
    __syncthreads();
    if (wid == 0) {
      float ss = wave_sum(red[lane]);
      if (lane == 0) {
        float lp = M + __logf(ss);
        if (i > 0) lp -= PENV;
        table[i] = lp;
      }
    }
    __syncthreads();
  }
  if (tid == 0) out[0] = table[0];
}

extern "C" void kernel_launch(void* const* d_in, const int* in_sizes, int n_in,
                              void* d_out, int out_size, void* d_ws, size_t ws_size,
                              hipStream_t stream) {
  const float* x       = (const float*)d_in[0];
  const int*   actions = (const int*)d_in[1];
  const float* micro_W = (const float*)d_in[2];
  const float* micro_b = (const float*)d_in[3];
  const float* sW1 = (const float*)d_in[4];
  const float* sb1 = (const float*)d_in[5];
  const float* sW2 = (const float*)d_in[6];
  const float* sb2 = (const float*)d_in[7];
  const float* sW3 = (const float*)d_in[8];
  const float* sb3 = (const float*)d_in[9];
  const float* gW1 = (const float*)d_in[10];
  const float* gb1 = (const float*)d_in[11];
  const float* gW2 = (const float*)d_in[12];
  const float* gb2 = (const float*)d_in[13];
  const float* gW3 = (const float*)d_in[14];
  const float* gb3 = (const float*)d_in[15];

  float* ws = (float*)d_ws;
  float* logits = ws;                 // TP1*4096; consumed before A is written
  float* Amat   = ws;                 // TLEN*TP1; overlays logits region
  size_t off = (size_t)TP1 * 4096;
  float* sH1 = ws + off;  off += (size_t)TP1 * 64;
  float* sH2 = ws + off;  off += (size_t)TP1 * 64;
  float* sZ  = ws + off;  off += (size_t)TP1 * 64;
  float* gH1 = ws + off;  off += (size_t)TP1 * 64;
  float* gH2 = ws + off;  off += (size_t)TP1 * 64;
  float* gZ  = ws + off;  off += (size_t)TP1 * 32;
  float* stepl   = ws + off;  off += (size_t)TLEN * 32;
  float* stop0   = ws + off;  off += (size_t)TP1 * 32;
  float* stop1   = ws + off;  off += (size_t)TP1 * 32;
  float* startlp = ws + off;  off += (size_t)TP1 * 32;
  float* Scum    = ws + off;  off += (size_t)TP1 * 32;
  float* C0      = ws + off;  off += (size_t)TP1 * 32;
  float* Lbuf    = ws + off;  off += (size_t)TLEN * 32;
  float* Rbuf    = ws + off;  off += (size_t)TP1 * 32;

  const dim3 gblk(256);
  const int mt32 = (TP1 + 31) / 32;  // 33

  // action logits: 1025x4096 = x @ micro_W + b
  gemm_wmma_f32<<<dim3((NBLK * NACT) / 64, mt32), gblk, 0, stream>>>(
      x, micro_W, micro_b, logits, TP1, NBLK * NACT, SDIM, 0);
  // stop MLP
  gemm_wmma_f32<<<dim3(1, mt32), gblk, 0, stream>>>(x,   sW1, sb1, sH1, TP1, HIDD, SDIM, 1);
  gemm_wmma_f32<<<dim3(1, mt32), gblk, 0, stream>>>(sH1, sW2, sb2, sH2, TP1, HIDD, HIDD, 1);
  gemm_wmma_f32<<<dim3(1, mt32), gblk, 0, stream>>>(sH2, sW3, sb3, sZ,  TP1, 2 * NBLK, HIDD, 0);
  // start MLP
  gemm_wmma_f32<<<dim3(1, mt32), gblk, 0, stream>>>(x,   gW1, gb1, gH1, TP1, HIDD, SDIM, 1);
  gemm_wmma_f32<<<dim3(1, mt32), gblk, 0, stream>>>(gH1, gW2, gb2, gH2, TP1, HIDD, HIDD, 1);
  gemm_wmma_f32<<<dim3(1, mt32), gblk, 0, stream>>>(gH2, gW3, gb3, gZ,  TP1, NBLK, HIDD, 0);

  step_logps_kernel<<<(TLEN * NBLK) / 4, dim3(128), 0, stream>>>(logits, actions, stepl);
  heads_kernel<<<(TP1 + 3) / 4, dim3(128), 0, stream>>>(sZ, gZ, stop0, stop1, startlp);
  scan_kernel<<<1, 32, 0, stream>>>(stepl, stop0, stop1, startlp, Scum, C0, Lbuf, Rbuf);
  pairA_kernel<<<dim3((TP1 + 15) / 16, TLEN / 16), 256, 0, stream>>>(Lbuf, Rbuf, Amat);
  dp_kernel<<<1, 1024, 0, stream>>>(Amat, (float*)d_out);
}